// RubikModel_20117626814600
// MI455X (gfx1250) — compile-verified
//
#include <hip/hip_runtime.h>
#include <math.h>

#define NUM_TBLOCKS 8
#define SEQ 54
#define DIM 6
#define QKV (3 * DIM)
#define FEAT (SEQ * DIM) /* 324 */
#define POL 12
#define LN_EPS 1e-5f
#define QSCALE 0.40824829046386302f /* 1/sqrt(6) */

#define WAVES_PER_BLOCK 4
#define THREADS (WAVES_PER_BLOCK * 32)
#define WAVE_LDS 3136 /* floats/wave: 3016 used + padded dump region */
#define WCACHE 704    /* block-shared padded weight cache (floats) */

typedef __attribute__((ext_vector_type(2))) float v2f;
typedef __attribute__((ext_vector_type(4))) float v4f;
typedef __attribute__((ext_vector_type(8))) float v8f;

// D = A(16x4 f32) * B(4x16 f32) + C(16x16 f32)
__device__ __forceinline__ v8f wmma_f32(v2f a, v2f b, v8f c) {
  return __builtin_amdgcn_wmma_f32_16x16x4_f32(false, a, false, b, (short)0, c,
                                               false, false);
}

__device__ __forceinline__ float eluf(float x) {
  return x > 0.0f ? x : (__expf(x) - 1.0f);
}

__global__ __launch_bounds__(THREADS) void rubik_fused(
    const float* __restrict__ x, const float* __restrict__ embed_w,
    const float* __restrict__ embed_b, const float* __restrict__ fc1_w,
    const float* __restrict__ fc1_b, const float* __restrict__ ln_g,
    const float* __restrict__ ln_b, const float* __restrict__ iw,
    const float* __restrict__ ib, const float* __restrict__ ow,
    const float* __restrict__ ob, const float* __restrict__ vw,
    const float* __restrict__ vb, const float* __restrict__ pw,
    const float* __restrict__ pb, float* __restrict__ out, int n_batch) {
  __shared__ float smem[WAVES_PER_BLOCK * WAVE_LDS + WCACHE];
  const int lane = threadIdx.x & 31;
  const int wave = threadIdx.x >> 5;
  int b = blockIdx.x * WAVES_PER_BLOCK + wave;
  if (b >= n_batch) b = n_batch - 1; // duplicate work writes identical values

  float* H = &smem[wave * WAVE_LDS];  // [54][6] running hidden state
  float* Z = H + 324;                 // [54][6] scratch (x-stage / z / attn-out)
  float* Qp = Z + 324;                // [64][8] zero-padded Q (pre-scaled)
  float* Kp = Qp + 512;               // [64][8] zero-padded K
  float* Vp = Kp + 512;               // [56][8] zero-padded V
  float* P = Vp + 448;                // [16][56] probs for current q-tile
  float* dump = H + 3016;             // per-wave discard region (16B aligned,
                                      // strided writes stay < WAVE_LDS)

  // block-shared padded weight cache
  float* WC = &smem[WAVES_PER_BLOCK * WAVE_LDS];
  float* W_FC = WC;        // 128: fc1 (or embed) B image
  float* W_OW = WC + 128;  // 128: out-proj B image
  float* W_Q0 = WC + 256;  // 128: qkv N-tile0 B image
  float* W_Q1 = WC + 384;  // 128: qkv N-tile1 B image
  float* B_FC = WC + 512;  // 32
  float* B_OW = WC + 544;  // 32
  float* B_Q0 = WC + 576;  // 32
  float* B_Q1 = WC + 608;  // 32

  const float* xb = x + (size_t)b * FEAT;
  __builtin_prefetch(xb, 0, 1); // global_prefetch_b8

  const int nloc = lane & 15;
  const int kb = (lane < 16) ? 0 : 2; // f32 A/B layout: lanes16-31 carry K=2,3
  const int upper8 = (lane < 16) ? 0 : 8;
  const bool lowhalf = (lane < 16);

  // qkv tile0 scatter constants (pure address arithmetic per lane)
  const int qgrp = nloc / 6;               // 0=Q, 1=K, 2=V
  const int qcol = nloc - qgrp * 6;        // column within Q/K/V
  const float qscl = (qgrp == 0) ? QSCALE : 1.0f;
  float* qkvBase = Qp + qgrp * 512 + qcol; // + row*8 at store time

  // Cooperative fill of one padded B image + bias image.
  // Image layout: wimg[kk*64 + lane*2 + p] = B[k = kk*4 + kb(lane) + p][n(lane)]
  auto fillWB = [&](const float* w, const float* bias, int nCols, int ntile,
                    float* wimg, float* bimg) {
    for (int slot = threadIdx.x; slot < 128; slot += THREADS) {
      int kk = slot >> 6, ln = (slot >> 1) & 31, p = slot & 1;
      int nl = ln & 15;
      int c = kk * 4 + ((ln < 16) ? 0 : 2) + p;
      int j = ntile * 16 + nl;
      bool v = (j < nCols) && (c < DIM);
      wimg[slot] = v ? w[v ? j * DIM + c : 0] : 0.0f;
    }
    for (int ln = threadIdx.x; ln < 32; ln += THREADS) {
      int j = ntile * 16 + (ln & 15);
      bimg[ln] = (j < nCols) ? bias[(j < nCols) ? j : 0] : 0.0f;
    }
  };

  // A tile (rows mt*16+nloc) from a [54][6] LDS matrix as aligned b64 loads;
  // reads may run into adjacent wave-local scratch (zeroed B rows nullify).
  auto loadA = [&](const float* src, int mt, v2f aw[2]) {
    int row = mt * 16 + nloc;
#pragma unroll
    for (int kk = 0; kk < 2; ++kk)
      aw[kk] = *(const v2f*)(src + row * DIM + kk * 4 + kb);
  };
  auto loadBimg = [&](const float* wimg, v2f bw[2]) {
    bw[0] = *(const v2f*)(wimg + lane * 2);
    bw[1] = *(const v2f*)(wimg + 64 + lane * 2);
  };
  auto loadCimg = [&](const float* bimg) {
    float bvv = bimg[lane];
    v8f c;
#pragma unroll
    for (int r = 0; r < 8; ++r) c[r] = bvv;
    return c;
  };

  // ---- stage x through LDS; zero Q/K/V pads once; stage embed weights ----
  for (int idx = lane; idx < FEAT; idx += 32) Z[idx] = xb[idx];
  for (int idx = lane; idx < 512 + 512 + 448; idx += 32) Qp[idx] = 0.0f;
  fillWB(embed_w, embed_b, DIM, 0, W_FC, B_FC);
  __syncthreads();

  // ---- embed: H = x @ embed_w^T + embed_b ----
  {
    v2f bw[2];
    loadBimg(W_FC, bw);
    v8f cb = loadCimg(B_FC);
#pragma unroll
    for (int mt = 0; mt < 4; ++mt) {
      v2f aw[2];
      loadA(Z, mt, aw);
      v8f d = wmma_f32(aw[0], bw[0], cb);
      d = wmma_f32(aw[1], bw[1], d);
      int row0 = mt * 16 + upper8;
      bool ok = (nloc < DIM) && ((mt < 3) || lowhalf);
      float* bz = ok ? &H[row0 * DIM + nloc] : dump;
#pragma unroll
      for (int r = 0; r < 8; ++r)
        if (mt < 3 || r < 6) bz[r * DIM] = d[r];
    }
  }
  __syncthreads();

  for (int L = 0; L < NUM_TBLOCKS; ++L) {
    const float* g = ln_g + L * DIM;
    const float* bbia = ln_b + L * DIM;

    // ---- stage this layer's padded weights (block-cooperative) ----
    fillWB(fc1_w + L * DIM * DIM, fc1_b + L * DIM, DIM, 0, W_FC, B_FC);
    fillWB(ow + L * DIM * DIM, ob + L * DIM, DIM, 0, W_OW, B_OW);
    fillWB(iw + L * QKV * DIM, ib + L * QKV, QKV, 0, W_Q0, B_Q0);
    fillWB(iw + L * QKV * DIM, ib + L * QKV, QKV, 1, W_Q1, B_Q1);
    __syncthreads();

    // ---- fc1 + ELU : Z = elu(H @ fw^T + fb) ----
    {
      v2f bw[2];
      loadBimg(W_FC, bw);
      v8f cb = loadCimg(B_FC);
#pragma unroll
      for (int mt = 0; mt < 4; ++mt) {
        v2f aw[2];
        loadA(H, mt, aw);
        v8f d = wmma_f32(aw[0], bw[0], cb);
        d = wmma_f32(aw[1], bw[1], d);
        int row0 = mt * 16 + upper8;
        bool ok = (nloc < DIM) && ((mt < 3) || lowhalf);
        float* bz = ok ? &Z[row0 * DIM + nloc] : dump;
#pragma unroll
        for (int r = 0; r < 8; ++r)
          if (mt < 3 || r < 6) bz[r * DIM] = eluf(d[r]);
      }
    }
    __syncthreads();

    // ---- LayerNorm over DIM (biased variance) ----
    for (int s = lane; s < SEQ; s += 32) {
      float mu = 0.0f;
#pragma unroll
      for (int i = 0; i < DIM; ++i) mu += Z[s * DIM + i];
      mu *= (1.0f / DIM);
      float var = 0.0f;
#pragma unroll
      for (int i = 0; i < DIM; ++i) {
        float d = Z[s * DIM + i] - mu;
        var += d * d;
      }
      var *= (1.0f / DIM);
      float rs = rsqrtf(var + LN_EPS);
#pragma unroll
      for (int i = 0; i < DIM; ++i)
        Z[s * DIM + i] = (Z[s * DIM + i] - mu) * rs * g[i] + bbia[i];
    }
    __syncthreads();

    // ---- packed QKV projection via WMMA (N=18 -> 2 N-tiles) ----
    {
      v2f bw0[2], bw1[2];
      loadBimg(W_Q0, bw0);
      loadBimg(W_Q1, bw1);
      v8f cb0 = loadCimg(B_Q0);
      v8f cb1 = loadCimg(B_Q1);
#pragma unroll
      for (int mt = 0; mt < 4; ++mt) {
        v2f aw[2];
        loadA(Z, mt, aw);
        v8f d0 = wmma_f32(aw[0], bw0[0], cb0);
        d0 = wmma_f32(aw[1], bw0[1], d0);
        v8f d1 = wmma_f32(aw[0], bw1[0], cb1);
        d1 = wmma_f32(aw[1], bw1[1], d1);
        int row0 = mt * 16 + upper8;
        bool ok0 = (mt < 3) || lowhalf;
        float* b0 = ok0 ? (qkvBase + row0 * 8) : dump;
        float* b1 = (ok0 && nloc < 2) ? (Vp + row0 * 8 + 4 + nloc) : dump;
#pragma unroll
        for (int r = 0; r < 8; ++r)
          if (mt < 3 || r < 6) {
            b0[r * 8] = d0[r] * qscl;
            b1[r * 8] = d1[r];
          }
      }
    }
    __syncthreads();

    // ---- hoist loop-invariant WMMA operands for attention ----
    v2f ak[4][2];
#pragma unroll
    for (int mt = 0; mt < 4; ++mt)
#pragma unroll
      for (int kk = 0; kk < 2; ++kk)
        ak[mt][kk] = *(const v2f*)(Kp + (mt * 16 + nloc) * 8 + kk * 4 + kb);
    v2f bv[14];
#pragma unroll
    for (int kc = 0; kc < 14; ++kc) {
      int kr = kc * 4 + kb;
      bv[kc].x = (nloc < 8) ? Vp[kr * 8 + nloc] : 0.0f;
      bv[kc].y = (nloc < 8) ? Vp[(kr + 1) * 8 + nloc] : 0.0f;
    }

    // ---- attention: scores TRANSPOSED (T = K @ Q^T); key-softmax is
    //      in-register + one xor-16 shuffle ----
    for (int nt = 0; nt < 4; ++nt) {
      v2f bq[2];
#pragma unroll
      for (int kk = 0; kk < 2; ++kk)
        bq[kk] = *(const v2f*)(Qp + (nt * 16 + nloc) * 8 + kk * 4 + kb);
      v8f t[4]; // t[mt][r] = score(key = mt*16 + r + upper8, query-lane)
#pragma unroll
      for (int mt = 0; mt < 4; ++mt) {
        v8f c = {};
#pragma unroll
        for (int kk = 0; kk < 2; ++kk) c = wmma_f32(ak[mt][kk], bq[kk], c);
        t[mt] = c;
      }

      float mx = -3.0e38f;
#pragma unroll
      for (int mt = 0; mt < 4; ++mt)
#pragma unroll
        for (int r = 0; r < 8; ++r) {
          int k = mt * 16 + r + upper8;
          if (k < SEQ) mx = fmaxf(mx, t[mt][r]);
        }
      mx = fmaxf(mx, __shfl_xor(mx, 16));
      float sum = 0.0f;
#pragma unroll
      for (int mt = 0; mt < 4; ++mt)
#pragma unroll
        for (int r = 0; r < 8; ++r) {
          int k = mt * 16 + r + upper8;
          float e = (k < SEQ) ? __expf(t[mt][r] - mx) : 0.0f;
          t[mt][r] = e;
          sum += e;
        }
      sum += __shfl_xor(sum, 16);
      float inv = 1.0f / sum;

      // P row segments are contiguous in k: two aligned b128 stores per tile
#pragma unroll
      for (int mt = 0; mt < 4; ++mt) {
        v4f lo = {t[mt][0] * inv, t[mt][1] * inv, t[mt][2] * inv,
                  t[mt][3] * inv};
        v4f hi = {t[mt][4] * inv, t[mt][5] * inv, t[mt][6] * inv,
                  t[mt][7] * inv};
        bool ok = (mt < 3) || lowhalf; // mt==3 upper half -> k >= 56
        float* bp = ok ? (P + nloc * 56 + mt * 16 + upper8) : dump;
        *(v4f*)(bp) = lo;
        *(v4f*)(bp + 4) = hi;
      }
      __syncthreads();

      // ---- O tile = P(16x56) @ V(56x6) via 14 K-steps ----
      v8f o = {};
#pragma unroll
      for (int kc = 0; kc < 14; ++kc) {
        v2f ap = *(const v2f*)(P + nloc * 56 + kc * 4 + kb);
        o = wmma_f32(ap, bv[kc], o);
      }
      {
        int row0 = nt * 16 + upper8;
        bool ok = (nloc < DIM) && ((nt < 3) || lowhalf);
        float* bz = ok ? &Z[row0 * DIM + nloc] : dump;
#pragma unroll
        for (int r = 0; r < 8; ++r)
          if (nt < 3 || r < 6) bz[r * DIM] = o[r];
      }
      __syncthreads();
    }

    // ---- out-proj + ELU + residual : H += elu(Z @ ow^T + ob) ----
    {
      v2f bw[2];
      loadBimg(W_OW, bw);
      v8f cb = loadCimg(B_OW);
#pragma unroll
      for (int mt = 0; mt < 4; ++mt) {
        v2f aw[2];
        loadA(Z, mt, aw);
        v8f d = wmma_f32(aw[0], bw[0], cb);
        d = wmma_f32(aw[1], bw[1], d);
        int row0 = mt * 16 + upper8;
        bool ok = (nloc < DIM) && ((mt < 3) || lowhalf);
        float* bh = ok ? &H[row0 * DIM + nloc] : dump;
#pragma unroll
        for (int r = 0; r < 8; ++r)
          if (mt < 3 || r < 6) {
            float hv = H[row0 * DIM + nloc + r * DIM]; // in-bounds of slice
            bh[r * DIM] = eluf(d[r]) + hv;
          }
      }
    }
    __syncthreads();
  }

  // ---- heads: value (1x324) and policy (12x324) + softmax ----
  float vacc = 0.0f;
  float pacc[POL];
#pragma unroll
  for (int o = 0; o < POL; ++o) pacc[o] = 0.0f;
  for (int j = lane; j < FEAT; j += 32) {
    float fj = H[j];
    vacc += fj * vw[j];
#pragma unroll
    for (int o = 0; o < POL; ++o) pacc[o] += fj * pw[o * FEAT + j];
  }
  for (int off = 16; off >= 1; off >>= 1) {
    vacc += __shfl_xor(vacc, off, 32);
#pragma unroll
    for (int o = 0; o < POL; ++o) pacc[o] += __shfl_xor(pacc[o], off, 32);
  }
  if (lane == 0) {
    out[b] = vacc + vb[0];
    float logit[POL];
    float mx = -3.0e38f;
#pragma unroll
    for (int o = 0; o < POL; ++o) {
      logit[o] = pacc[o] + pb[o];
      mx = fmaxf(mx, logit[o]);
    }
    float sum = 0.0f;
#pragma unroll
    for (int o = 0; o < POL; ++o) {
      logit[o] = __expf(logit[o] - mx);
      sum += logit[o];
    }
    float inv = 1.0f / sum;
    float* pout = out + n_batch + (size_t)b * POL;
#pragma unroll
    for (int o = 0; o < POL; ++o) pout[o] = logit[o] * inv;
  }
}

extern "C" void kernel_launch(void* const* d_in, const int* in_sizes, int n_in,
                              void* d_out, int out_size, void* d_ws,
                              size_t ws_size, hipStream_t stream) {
  (void)n_in;
  (void)out_size;
  (void)d_ws;
  (void)ws_size;
  const float* x = (const float*)d_in[0];
  const float* embed_w = (const float*)d_in[1];
  const float* embed_b = (const float*)d_in[2];
  const float* fc1_w = (const float*)d_in[3];
  const float* fc1_b = (const float*)d_in[4];
  const float* ln_g = (const float*)d_in[5];
  const float* ln_b = (const float*)d_in[6];
  const float* iw = (const float*)d_in[7];
  const float* ib = (const float*)d_in[8];
  const float* ow = (const float*)d_in[9];
  const float* ob = (const float*)d_in[10];
  const float* vw = (const float*)d_in[11];
  const float* vb = (const float*)d_in[12];
  const float* pw = (const float*)d_in[13];
  const float* pb = (const float*)d_in[14];

  int n_batch = in_sizes[0] / FEAT;
  int grid = (n_batch + WAVES_PER_BLOCK - 1) / WAVES_PER_BLOCK;
  rubik_fused<<<grid, THREADS, 0, stream>>>(x, embed_w, embed_b, fc1_w, fc1_b,
                                            ln_g, ln_b, iw, ib, ow, ob, vw, vb,
                                            pw, pb, (float*)d_out, n_batch);
}